// VectorQuantizationIndexes_21088289423520
// MI455X (gfx1250) — compile-verified
//
#include <hip/hip_runtime.h>

typedef __attribute__((ext_vector_type(2))) float v2f;
typedef __attribute__((ext_vector_type(4))) float v4f;
typedef __attribute__((ext_vector_type(8))) float v8f;

#define K_CODES   2048
#define C_DIM     4
#define TILE_K    16
#define NUM_TILES (K_CODES / TILE_K)   // 128
#define BLOCK_THREADS 256              // 8 wave32 waves
#define ROWS_PER_WAVE 16

// score[n,k] = ||e_k||^2 - 2*x_n.e_k  ==  e2[k] + x_n.(-2*e_k)
// One V_WMMA_F32_16X16X4_F32 computes a full 16x16 score tile:
//   A (16x4)  = 16 rows of x              (2 VGPRs / lane)
//   B (4x16)  = (-2*emb) columns k..k+15  (2 VGPRs / lane, from LDS)
//   C (16x16) = e2[k] broadcast per column (v8f splat; C's N dim == lane&15)
__global__ __launch_bounds__(BLOCK_THREADS)
void vq_argmin_wmma_kernel(const float* __restrict__ x,
                           const float* __restrict__ emb,
                           int* __restrict__ out,
                           int n_total)
{
    __shared__ float lds_b [K_CODES * C_DIM]; // -2 * embedding, row major [k][c]
    __shared__ float lds_e2[K_CODES];         // ||e_k||^2

    const int tid = threadIdx.x;

    // Cooperative stage: 256 threads x 8 codes each.
    for (int k = tid; k < K_CODES; k += BLOCK_THREADS) {
        v4f e = *(const v4f*)(emb + (size_t)k * C_DIM);
        lds_e2[k] = e.x * e.x + e.y * e.y + e.z * e.z + e.w * e.w;
        v4f b = { -2.0f * e.x, -2.0f * e.y, -2.0f * e.z, -2.0f * e.w };
        *(v4f*)(lds_b + (size_t)k * C_DIM) = b;
    }
    __syncthreads();

    const int lane    = tid & 31;
    const int halfSel = lane >> 4;      // 0: lanes 0-15 (K=0,1 / M=0-7), 1: lanes 16-31 (K=2,3 / M=8-15)
    const int l15     = lane & 15;
    const int koff    = halfSel * 2;    // which pair of K values this half-wave carries

    const int waveId = blockIdx.x * (BLOCK_THREADS / 32) + (tid >> 5);
    const int n_base = waveId * ROWS_PER_WAVE;
    if (n_base >= n_total) return;

    // A operand: 16 rows of x, loaded once, reused across all 128 WMMAs.
    // A layout (32-bit 16x4): lanes 0-15 -> M=lane, VGPR{0,1}=K{0,1}; lanes 16-31 -> K{2,3}.
    v2f a = *(const v2f*)(x + (size_t)(n_base + l15) * C_DIM + koff);

    float bestv[8];
    int   besti[8];
#pragma unroll
    for (int v = 0; v < 8; ++v) { bestv[v] = 3.4e38f; besti[v] = 0; }

    for (int t = 0; t < NUM_TILES; ++t) {
        const int kcol = t * TILE_K + l15;   // this lane's column (code index) in the tile

        // B layout (32-bit 4x16): lanes 0-15 -> N=lane, VGPR{0,1}=K{0,1}; lanes 16-31 -> K{2,3}.
        v2f b = *(const v2f*)(lds_b + (size_t)kcol * C_DIM + koff);

        const float e2 = lds_e2[kcol];
        v8f c = { e2, e2, e2, e2, e2, e2, e2, e2 };

        // (neg_a, A, neg_b, B, c_mod, C, reuse_a, reuse_b)
        c = __builtin_amdgcn_wmma_f32_16x16x4_f32(
                false, a, false, b, (short)0, c, false, false);

        // Running argmin over k for the 8 rows this lane sees (same column kcol for all 8).
        // Tiles ascend in k and '<' is strict -> first-minimum semantics like jnp.argmin.
#pragma unroll
        for (int v = 0; v < 8; ++v) {
            float s  = c[v];
            bool  lt = s < bestv[v];
            bestv[v] = lt ? s    : bestv[v];
            besti[v] = lt ? kcol : besti[v];
        }
    }

    // Columns of each tile were spread across 16 lanes: reduce (val, idx) pairs
    // with a butterfly over XOR masks 1,2,4,8 (stays inside each 16-lane half).
    // Value ties break toward the smaller code index (argmin first-occurrence).
#pragma unroll
    for (int sh = 1; sh <= 8; sh <<= 1) {
#pragma unroll
        for (int v = 0; v < 8; ++v) {
            float ov = __shfl_xor(bestv[v], sh, 32);
            int   oi = __shfl_xor(besti[v], sh, 32);
            bool  take = (ov < bestv[v]) || (ov == bestv[v] && oi < besti[v]);
            bestv[v] = take ? ov : bestv[v];
            besti[v] = take ? oi : besti[v];
        }
    }

    // C layout: lanes 0-15 hold rows 0-7 (VGPR v -> M=v), lanes 16-31 hold rows 8-15.
    // After the butterfly all 16 lanes in a half agree; lane (l15 == r, r<8) emits row r.
    const int r = lane & 7;
    int sel = besti[0];
#pragma unroll
    for (int v = 1; v < 8; ++v) sel = (r == v) ? besti[v] : sel;

    if (l15 < 8) {
        out[n_base + halfSel * 8 + r] = sel;
    }
}

extern "C" void kernel_launch(void* const* d_in, const int* in_sizes, int n_in,
                              void* d_out, int out_size, void* d_ws, size_t ws_size,
                              hipStream_t stream) {
    const float* x   = (const float*)d_in[0];   // [16,64,64,4] f32
    const float* emb = (const float*)d_in[1];   // [2048,4]     f32
    int* out = (int*)d_out;                     // [16,64,64]   int32 indices

    const int n_total = in_sizes[0] / C_DIM;    // 65536 rows
    const int waves   = (n_total + ROWS_PER_WAVE - 1) / ROWS_PER_WAVE;         // 4096
    const int blocks  = (waves + (BLOCK_THREADS / 32) - 1) / (BLOCK_THREADS / 32); // 512

    vq_argmin_wmma_kernel<<<blocks, BLOCK_THREADS, 0, stream>>>(x, emb, out, n_total);
}